// PAEmbedding_19018115187089
// MI455X (gfx1250) — compile-verified
//
#include <hip/hip_runtime.h>
#include <math.h>

// Problem constants (match reference)
#define Bn 32
#define Tn 512
#define Cn 512
#define Dm 512
#define NHh 4
#define HDh 128

typedef float v2f __attribute__((ext_vector_type(2)));
typedef float v8f __attribute__((ext_vector_type(8)));

// ---------------------------------------------------------------------------
// helpers
// ---------------------------------------------------------------------------
__device__ __forceinline__ int clampi(int x, int lo, int hi) {
  return x < lo ? lo : (x > hi ? hi : x);
}

// Bilinear resize row t of M[p, a]: two sparse entries (j0,w0),(j1,w1)
__device__ __forceinline__ void row_w(int t, int p, int a,
                                      int& j0, int& j1, float& w0, float& w1) {
  float src = ((float)t + 0.5f) * (float)a / (float)p - 0.5f;
  float fl  = floorf(src);
  int   i0  = (int)fl;
  float fr  = src - fl;
  j0 = clampi(i0, 0, a - 1);
  j1 = clampi(i0 + 1, 0, a - 1);
  w0 = 1.0f - fr;
  w1 = fr;
}

// Block-cooperative Gauss-Jordan on SPD matrix in aug[n][2n] (row stride 96).
// Result: inverse in columns [n, 2n).
__device__ void gauss_jordan(float* aug, float* fac, int n, int tid, int nthr) {
  for (int k = 0; k < n; ++k) {
    __syncthreads();
    float piv = aug[k * 96 + k];
    float ip  = 1.0f / piv;
    __syncthreads();
    for (int c2 = tid; c2 < 2 * n; c2 += nthr) aug[k * 96 + c2] *= ip;
    __syncthreads();
    for (int r = tid; r < n; r += nthr) fac[r] = (r == k) ? 0.0f : aug[r * 96 + k];
    __syncthreads();
    for (int idx = tid; idx < n * 2 * n; idx += nthr) {
      int r = idx / (2 * n), c2 = idx % (2 * n);
      aug[r * 96 + c2] -= fac[r] * aug[k * 96 + c2];
    }
  }
  __syncthreads();
}

// Build pinvT[j*p + t] = scale * pinv(M.T)[t, j]  (rows j >= a zeroed).
// pinvT may live in LDS or global; aug/fac are LDS scratch (48*96 / 64 floats).
__device__ void build_pinvT(float* pinvT, float* aug, float* fac,
                            int p, int a, int tid, int nthr) {
  float scale = sqrtf((float)p / (float)a);
  for (int idx = tid; idx < 48 * 96; idx += nthr) aug[idx] = 0.0f;
  __syncthreads();
  if (p >= a) {
    // pinv = M (M^T M)^-1 ; G = M^T M is a x a SPD
    for (int r = tid; r < a; r += nthr) aug[r * 96 + a + r] = 1.0f;
    __syncthreads();
    for (int t = tid; t < p; t += nthr) {
      int j0, j1; float w0, w1;
      row_w(t, p, a, j0, j1, w0, w1);
      if (j0 == j1) {
        atomicAdd(&aug[j0 * 96 + j0], (w0 + w1) * (w0 + w1));
      } else {
        atomicAdd(&aug[j0 * 96 + j0], w0 * w0);
        atomicAdd(&aug[j0 * 96 + j1], w0 * w1);
        atomicAdd(&aug[j1 * 96 + j0], w1 * w0);
        atomicAdd(&aug[j1 * 96 + j1], w1 * w1);
      }
    }
    __syncthreads();
    gauss_jordan(aug, fac, a, tid, nthr);
    for (int idx = tid; idx < 48 * p; idx += nthr) {
      int j = idx / p, t = idx % p;
      float v = 0.0f;
      if (j < a) {
        int j0, j1; float w0, w1;
        row_w(t, p, a, j0, j1, w0, w1);
        v = scale * (w0 * aug[j0 * 96 + a + j] + w1 * aug[j1 * 96 + a + j]);
      }
      pinvT[idx] = v;
    }
  } else {
    // pinv = (M M^T)^-1 M ; H = M M^T is p x p SPD (p <= 47)
    for (int r = tid; r < p; r += nthr) aug[r * 96 + p + r] = 1.0f;
    __syncthreads();
    for (int idx = tid; idx < p * p; idx += nthr) {
      int t = idx / p, u = idx % p;
      int a0, a1, b0, b1; float wa0, wa1, wb0, wb1;
      row_w(t, p, a, a0, a1, wa0, wa1);
      row_w(u, p, a, b0, b1, wb0, wb1);
      float v = 0.0f;
      if (a0 == b0) v += wa0 * wb0;
      if (a0 == b1) v += wa0 * wb1;
      if (a1 == b0) v += wa1 * wb0;
      if (a1 == b1) v += wa1 * wb1;
      aug[t * 96 + u] = v;
    }
    __syncthreads();
    gauss_jordan(aug, fac, p, tid, nthr);
    for (int idx = tid; idx < 48 * p; idx += nthr) {
      int j = idx / p, t = idx % p;
      float v = 0.0f;
      if (j < a) {
        for (int u = 0; u < p; ++u) {
          int j0, j1; float w0, w1;
          row_w(u, p, a, j0, j1, w0, w1);
          float m = ((j0 == j) ? w0 : 0.0f) + ((j1 == j) ? w1 : 0.0f);
          v += aug[t * 96 + p + u] * m;
        }
        v *= scale;
      }
      pinvT[idx] = v;
    }
  }
  __syncthreads();
}

__device__ __forceinline__ float gelu_exact(float x) {
  return 0.5f * x * (1.0f + erff(x * 0.7071067811865476f));
}

// ---------------------------------------------------------------------------
// 1) transpose x_s [B,T,C] -> x_inv [B*C, T]
// ---------------------------------------------------------------------------
__global__ void k_transpose(const float* __restrict__ xs, float* __restrict__ xinv) {
  __shared__ float tile[32][33];
  int b  = blockIdx.z;
  int t0 = blockIdx.x * 32;
  int c0 = blockIdx.y * 32;
  int tx = threadIdx.x, ty = threadIdx.y;
  for (int q = 0; q < 4; ++q) {
    int r = ty + q * 8;
    tile[r][tx] = xs[((size_t)b * Tn + (t0 + r)) * Cn + (c0 + tx)];
  }
  __syncthreads();
  for (int q = 0; q < 4; ++q) {
    int r = ty + q * 8;
    xinv[((size_t)b * Cn + (c0 + r)) * Tn + (t0 + tx)] = tile[tx][r];
  }
}

// ---------------------------------------------------------------------------
// 2) ACF + top-2 periods per channel (one WG per channel)
// ---------------------------------------------------------------------------
__global__ __launch_bounds__(256) void k_acf(const float* __restrict__ xinv,
                                             int* __restrict__ periods) {
  __shared__ float row[512];
  __shared__ float acf[512];
  __shared__ float red[256];
  int c = blockIdx.x, tid = threadIdx.x;
  int l0 = tid, l1 = tid + 256;
  double acc0 = 0.0, acc1 = 0.0;
  for (int b = 0; b < Bn; ++b) {
    __syncthreads();
    const float* src = xinv + ((size_t)b * Cn + c) * Tn;
    row[tid]       = src[tid];
    row[tid + 256] = src[tid + 256];
    __syncthreads();
    red[tid] = row[tid] + row[tid + 256];
    __syncthreads();
    for (int s = 128; s > 0; s >>= 1) {
      if (tid < s) red[tid] += red[tid + s];
      __syncthreads();
    }
    float mean = red[0] * (1.0f / 512.0f);
    __syncthreads();
    row[tid]       -= mean;
    row[tid + 256] -= mean;
    __syncthreads();
    double s0 = 0.0, s1 = 0.0;
    for (int t = 0; t + l0 < Tn; ++t) s0 += (double)(row[t] * row[t + l0]);
    for (int t = 0; t + l1 < Tn; ++t) s1 += (double)(row[t] * row[t + l1]);
    acc0 += s0;
    acc1 += s1;
  }
  __syncthreads();
  acf[l0] = (float)(acc0 / (double)Bn);
  acf[l1] = (float)(acc1 / (double)Bn);
  __syncthreads();
  float a0 = acf[0] + 1e-9f;
  __syncthreads();
  acf[l0] = (l0 < 4) ? -__builtin_inff() : acf[l0] / a0;
  acf[l1] = acf[l1] / a0;
  __syncthreads();
  if (tid == 0) {
    float b1v = -__builtin_inff(), b2v = -__builtin_inff();
    int   b1i = -1, b2i = -1;
    for (int l = 4; l < Tn; ++l) {
      float v    = acf[l];
      float prev = acf[l - 1];
      float next = (l == Tn - 1) ? acf[0] : acf[l + 1];  // np.roll wraps
      if (v > prev && v > next && v > 0.15f) {
        if (v > b1v) { b2v = b1v; b2i = b1i; b1v = v; b1i = l; }
        else if (v > b2v) { b2v = v; b2i = l; }
      }
    }
    periods[c * 2 + 0] = (b1i >= 0) ? b1i : Tn;
    periods[c * 2 + 1] = (b2i >= 0) ? b2i : Tn;
  }
}

// ---------------------------------------------------------------------------
// 3) pinvT for the trend path (p=512, a=48) -> global ws
// ---------------------------------------------------------------------------
__global__ __launch_bounds__(256) void k_pinv512(float* __restrict__ pinvT) {
  extern __shared__ float smem[];
  float* aug = smem;
  float* fac = smem + 48 * 96;
  build_pinvT(pinvT, aug, fac, 512, 48, threadIdx.x, 256);
}

// ---------------------------------------------------------------------------
// 4) generic fp32 WMMA GEMM:  C = act(A[M,K] @ W[N,K]^T + bias)
//    64x64 block tile, 8 waves, each wave: two 16x16 accumulators.
// ---------------------------------------------------------------------------
__global__ __launch_bounds__(256) void k_gemm(const float* __restrict__ A, int lda,
                                              const float* __restrict__ W, int ldw,
                                              const float* __restrict__ bias,
                                              float* __restrict__ Cp, int ldc,
                                              int M, int N, int K, int act) {
  __shared__ float As[64 * 17];
  __shared__ float Ws[64 * 17];
  int tid  = threadIdx.x;
  int m0   = blockIdx.x * 64;
  int n0   = blockIdx.y * 64;
  int wave = tid >> 5, lane = tid & 31;
  int wm = wave & 3, wn = wave >> 2;     // 4 M-subtiles x 2 N-halves
  int fm = lane & 15;
  int kh = (lane >> 4) * 2;              // K split across lane halves
  v8f c0 = {};
  v8f c1 = {};
  int lr = tid >> 2;                     // 0..63 staging row
  int lk = (tid & 3) * 4;                // 0,4,8,12 staging col

  for (int k0 = 0; k0 < K; k0 += 16) {
    __syncthreads();
    float4 av = *reinterpret_cast<const float4*>(A + (size_t)(m0 + lr) * lda + k0 + lk);
    As[lr * 17 + lk + 0] = av.x;
    As[lr * 17 + lk + 1] = av.y;
    As[lr * 17 + lk + 2] = av.z;
    As[lr * 17 + lk + 3] = av.w;
    int nr = n0 + lr;
    float4 wv4;
    if (nr < N) wv4 = *reinterpret_cast<const float4*>(W + (size_t)nr * ldw + k0 + lk);
    else { wv4.x = wv4.y = wv4.z = wv4.w = 0.0f; }
    Ws[lr * 17 + lk + 0] = wv4.x;
    Ws[lr * 17 + lk + 1] = wv4.y;
    Ws[lr * 17 + lk + 2] = wv4.z;
    Ws[lr * 17 + lk + 3] = wv4.w;
    if (k0 + 16 < K) {                   // global_prefetch_b8 of next K-tile
      __builtin_prefetch(A + (size_t)(m0 + lr) * lda + k0 + 16 + lk, 0, 1);
      if (nr < N) __builtin_prefetch(W + (size_t)nr * ldw + k0 + 16 + lk, 0, 1);
    }
    __syncthreads();
#pragma unroll
    for (int ks = 0; ks < 4; ++ks) {
      v2f af, bf0, bf1;
      af.x  = As[(wm * 16 + fm) * 17 + ks * 4 + kh];
      af.y  = As[(wm * 16 + fm) * 17 + ks * 4 + kh + 1];
      bf0.x = Ws[(wn * 32 + fm) * 17 + ks * 4 + kh];
      bf0.y = Ws[(wn * 32 + fm) * 17 + ks * 4 + kh + 1];
      bf1.x = Ws[(wn * 32 + 16 + fm) * 17 + ks * 4 + kh];
      bf1.y = Ws[(wn * 32 + 16 + fm) * 17 + ks * 4 + kh + 1];
      c0 = __builtin_amdgcn_wmma_f32_16x16x4_f32(false, af, false, bf0, (short)0, c0, false, false);
      c1 = __builtin_amdgcn_wmma_f32_16x16x4_f32(false, af, false, bf1, (short)0, c1, false, false);
    }
  }
  int mbase = m0 + wm * 16 + (lane >> 4) * 8;
  int nc0   = n0 + wn * 32 + fm;
  int nc1   = nc0 + 16;
#pragma unroll
  for (int r = 0; r < 8; ++r) {
    int gm = mbase + r;
    float v0 = c0[r], v1 = c1[r];
    if (bias) {
      if (nc0 < N) v0 += bias[nc0];
      if (nc1 < N) v1 += bias[nc1];
    }
    if (act == 1) { v0 = gelu_exact(v0); v1 = gelu_exact(v1); }
    if (nc0 < N) Cp[(size_t)gm * ldc + nc0] = v0;
    if (nc1 < N) Cp[(size_t)gm * ldc + nc1] = v1;
  }
}

// ---------------------------------------------------------------------------
// 5) fold attention weights through anchors
// WKEFF/WVEFF: [(i*3+ai)*512 + n]*48 + j  = (wk|wv) @ w_a
// ---------------------------------------------------------------------------
__global__ void k_weff(const float* __restrict__ inw1, const float* __restrict__ inw2,
                       const float* __restrict__ w12, const float* __restrict__ w24,
                       const float* __restrict__ w48,
                       float* __restrict__ WKEFF, float* __restrict__ WVEFF) {
  int idx = blockIdx.x * blockDim.x + threadIdx.x;
  int total = 2 * 3 * 512 * 48;
  if (idx >= total) return;
  int j  = idx % 48;
  int n  = (idx / 48) % 512;
  int ai = (idx / (48 * 512)) % 3;
  int i  = idx / (48 * 512 * 3);
  int a  = (ai == 0) ? 12 : ((ai == 1) ? 24 : 48);
  const float* wa  = (ai == 0) ? w12 : ((ai == 1) ? w24 : w48);
  const float* inw = i ? inw2 : inw1;
  const float* wk  = inw + 512 * 512;
  const float* wv  = inw + 1024 * 512;
  float sk = 0.0f, sv = 0.0f;
  if (j < a) {
    for (int k = 0; k < 512; ++k) {
      float wav = wa[k * a + j];
      sk += wk[n * 512 + k] * wav;
      sv += wv[n * 512 + k] * wav;
    }
  }
  WKEFF[idx] = sk;
  WVEFF[idx] = sv;
}

// bv_eff[i][j] = shared_bias @ wv_i^T + bv_i
__global__ void k_bveff(const float* __restrict__ sb,
                        const float* __restrict__ inw1, const float* __restrict__ inw2,
                        const float* __restrict__ inb1, const float* __restrict__ inb2,
                        float* __restrict__ BVEFF) {
  int idx = blockIdx.x * blockDim.x + threadIdx.x;
  if (idx >= 2 * 512) return;
  int i = idx / 512, j = idx % 512;
  const float* wv = (i ? inw2 : inw1) + 1024 * 512;
  const float* bv = (i ? inb2 : inb1) + 1024;
  float s = bv[j];
  for (int k = 0; k < 512; ++k) s += sb[k] * wv[j * 512 + k];
  BVEFF[idx] = s;
}

// WQK[(tb*512+k)*192+col] = sum_r wq[h*128+r, k] * WKEFF[tb][h*128+r][j]
// WCTX[(tb*192+col)*512+n] = sum_r out_w[n, h*128+r] * WVEFF[tb][h*128+r][j]
__global__ void k_wqk_wctx(const float* __restrict__ inw1, const float* __restrict__ inw2,
                           const float* __restrict__ ow1, const float* __restrict__ ow2,
                           const float* __restrict__ WKEFF, const float* __restrict__ WVEFF,
                           float* __restrict__ WQK, float* __restrict__ WCTX) {
  int idx = blockIdx.x * blockDim.x + threadIdx.x;
  int total = 2 * 3 * 512 * 192;
  if (idx >= total) return;
  int col = idx % 192;
  int k   = (idx / 192) % 512;
  int ai  = (idx / (192 * 512)) % 3;
  int i   = idx / (192 * 512 * 3);
  int tb  = i * 3 + ai;
  int h = col / 48, j = col % 48;
  int a = (ai == 0) ? 12 : ((ai == 1) ? 24 : 48);
  const float* wq = i ? inw2 : inw1;  // rows 0..511
  const float* ow = i ? ow2 : ow1;
  float sq = 0.0f, sc2 = 0.0f;
  if (j < a) {
    for (int r = 0; r < 128; ++r) {
      int hr = h * 128 + r;
      float ke = WKEFF[((size_t)tb * 512 + hr) * 48 + j];
      float ve = WVEFF[((size_t)tb * 512 + hr) * 48 + j];
      sq  += wq[(size_t)hr * 512 + k] * ke;
      sc2 += ow[(size_t)k * 512 + hr] * ve;
    }
  }
  WQK[idx] = sq;
  WCTX[((size_t)tb * 192 + col) * 512 + k] = sc2;
}

// BQK[tb*192+col] = sum_r bq[h*128+r]*WKEFF[..] ; BOUT[i*512+n] = out_w@bv_eff + out_b
__global__ void k_bqk_bout(const float* __restrict__ inb1, const float* __restrict__ inb2,
                           const float* __restrict__ ow1, const float* __restrict__ ow2,
                           const float* __restrict__ ob1, const float* __restrict__ ob2,
                           const float* __restrict__ WKEFF, const float* __restrict__ BVEFF,
                           float* __restrict__ BQK, float* __restrict__ BOUT) {
  int idx = blockIdx.x * blockDim.x + threadIdx.x;
  if (idx < 2 * 3 * 192) {
    int col = idx % 192, tb = idx / 192;
    int i = tb / 3, ai = tb % 3;
    int h = col / 48, j = col % 48;
    int a = (ai == 0) ? 12 : ((ai == 1) ? 24 : 48);
    const float* bq = i ? inb2 : inb1;  // rows 0..511
    float s = 0.0f;
    if (j < a)
      for (int r = 0; r < 128; ++r)
        s += bq[h * 128 + r] * WKEFF[((size_t)tb * 512 + h * 128 + r) * 48 + j];
    BQK[idx] = s;
  } else if (idx < 2 * 3 * 192 + 2 * 512) {
    int idx2 = idx - 2 * 3 * 192;
    int i = idx2 / 512, n = idx2 % 512;
    const float* ow = i ? ow2 : ow1;
    const float* ob = i ? ob2 : ob1;
    float s = ob[n];
    for (int j = 0; j < 512; ++j) s += ow[(size_t)n * 512 + j] * BVEFF[i * 512 + j];
    BOUT[idx2] = s;
  }
}

// ---------------------------------------------------------------------------
// 6) per (stream, channel) aggregation: pinv -> proj -> attention -> LN
// ---------------------------------------------------------------------------
__global__ __launch_bounds__(256) void k_aggregate(
    const float* __restrict__ xinv, const int* __restrict__ periods,
    const float* __restrict__ w12, const float* __restrict__ w24,
    const float* __restrict__ w48, const float* __restrict__ sb,
    const float* __restrict__ WQK, const float* __restrict__ BQK,
    const float* __restrict__ WCTX, const float* __restrict__ BOUT,
    const float* __restrict__ ln1g, const float* __restrict__ ln1b,
    const float* __restrict__ ln2g, const float* __restrict__ ln2b,
    float* __restrict__ combined) {
  extern __shared__ float smem[];
  float* pinvT = smem;                 // 24576
  float* aug   = pinvT + 24576;        // 4608
  float* fac   = aug + 4608;           // 64
  float* xrow  = fac + 64;             // 512
  float* proj  = xrow + 512;           // 6144 (128 x 48)
  float* qin   = proj + 6144;          // 512
  float* resv  = qin + 512;            // 512
  float* qk    = resv + 512;           // 192
  float* ctx   = qk + 192;             // 192
  float* pm    = ctx + 192;            // 64
  float* sc    = pm + 64;              // 512 (4 x 128)
  float* red   = sc + 512;             // 256

  int c   = blockIdx.x;
  int i   = blockIdx.y;
  int tid = threadIdx.x;
  int p   = periods[c * 2 + i];

  if (p >= Tn) {  // channel not selected for this stream -> zeros
    for (int idx = tid; idx < Bn * Dm; idx += 256) {
      int b = idx / Dm, n = idx % Dm;
      combined[((size_t)b * Cn + c) * (3 * Dm) + i * Dm + n] = 0.0f;
    }
    return;
  }

  int a  = (p <= 18) ? 12 : ((p <= 36) ? 24 : 48);
  int ai = (a == 12) ? 0 : ((a == 24) ? 1 : 2);
  int tb = i * 3 + ai;
  const float* wa  = (ai == 0) ? w12 : ((ai == 1) ? w24 : w48);
  const float* lng = i ? ln2g : ln1g;
  const float* lnb = i ? ln2b : ln1b;

  build_pinvT(pinvT, aug, fac, p, a, tid, 256);

  int L  = Tn / p;
  int Lp = L * p;
  const float inv_sqrt_hd = 0.08838834764831845f;  // 1/sqrt(128)

  for (int b = 0; b < Bn; ++b) {
    __syncthreads();
    const float* src = xinv + ((size_t)b * Cn + c) * Tn;
    for (int t = tid; t < Lp; t += 256) xrow[t] = src[t];
    __syncthreads();

    // proj[l, j] = sum_t xrow[l*p+t] * pinvT[j*p+t]   (rows j>=a are zero)
    for (int idx = tid; idx < L * 48; idx += 256) {
      int l = idx / 48, j = idx % 48;
      const float* pj = pinvT + (size_t)j * p;
      const float* xr = xrow + l * p;
      float v = 0.0f;
      for (int t = 0; t < p; ++t) v += xr[t] * pj[t];
      proj[idx] = v;
    }
    __syncthreads();

    // query seed: mean of last two patches (or last if L==1)
    for (int j = tid; j < 48; j += 256)
      pm[j] = (L >= 2) ? 0.5f * (proj[(L - 1) * 48 + j] + proj[(L - 2) * 48 + j])
                       : proj[j];
    __syncthreads();

    // q_in = pm @ w_a^T + shared_bias  (full 512-dim; also the residual)
    for (int n = tid; n < Dm; n += 256) {
      float v = sb[n];
      for (int j = 0; j < a; ++j) v += pm[j] * wa[n * a + j];
      qin[n] = v;
    }
    __syncthreads();

    // qk[col] = q_in @ WQK + BQK (folded q- and k-projections)
    for (int col = tid; col < 192; col += 256) {
      float v = BQK[tb * 192 + col];
      const float* wq = WQK + (size_t)tb * 512 * 192 + col;
      for (int k = 0; k < 512; ++k) v += qin[k] * wq[(size_t)k * 192];
      qk[col] = v;
    }
    __syncthreads();

    // scores
    for (int idx = tid; idx < NHh * L; idx += 256) {
      int h = idx / L, l = idx % L;
      float v = 0.0f;
      for (int j = 0; j < 48; ++j) v += qk[h * 48 + j] * proj[l * 48 + j];
      sc[h * L + l] = v * inv_sqrt_hd;
    }
    __syncthreads();

    // softmax per head (L <= 128)
    if (tid < NHh) {
      float mx = -__builtin_inff();
      for (int l = 0; l < L; ++l) mx = fmaxf(mx, sc[tid * L + l]);
      float ssum = 0.0f;
      for (int l = 0; l < L; ++l) {
        float e = expf(sc[tid * L + l] - mx);
        sc[tid * L + l] = e;
        ssum += e;
      }
      float inv = 1.0f / ssum;
      for (int l = 0; l < L; ++l) sc[tid * L + l] *= inv;
    }
    __syncthreads();

    // ctx[h, j] = attn @ proj
    for (int idx = tid; idx < 192; idx += 256) {
      int h = idx / 48, j = idx % 48;
      float v = 0.0f;
      for (int l = 0; l < L; ++l) v += sc[h * L + l] * proj[l * 48 + j];
      ctx[idx] = v;
    }
    __syncthreads();

    // out = ctx @ WCTX + BOUT ; residual
    for (int n = tid; n < Dm; n += 256) {
      float v = BOUT[i * Dm + n];
      for (int col = 0; col < 192; ++col)
        v += ctx[col] * WCTX[((size_t)tb * 192 + col) * 512 + n];
      resv[n] = qin[n] + v;
    }
    __syncthreads();

    // LayerNorm
    red[tid] = resv[tid] + resv[tid + 256];
    __syncthreads();
    for (int s = 128; s > 0; s >>= 1) {
      if (tid < s) red[tid] += red[tid + s];
      __syncthreads();
    }
    float mean = red[0] * (1.0f / 512.0f);
    __syncthreads();
    float d0 = resv[tid] - mean, d1 = resv[tid + 256] - mean;
    red[tid] = d0 * d0 + d1 * d1;
    __syncthreads();
    for (int s = 128; s > 0; s >>= 1) {
      if (tid < s) red[tid] += red[tid + s];
      __syncthreads();
    }
    float rstd = rsqrtf(red[0] * (1.0f / 512.0f) + 1e-5f);
    __syncthreads();
    for (int n = tid; n < Dm; n += 256) {
      float y = (resv[n] - mean) * rstd * lng[n] + lnb[n];
      combined[((size_t)b * Cn + c) * (3 * Dm) + i * Dm + n] = y;
    }
  }
}

// ---------------------------------------------------------------------------
// launch
// ---------------------------------------------------------------------------
extern "C" void kernel_launch(void* const* d_in, const int* in_sizes, int n_in,
                              void* d_out, int out_size, void* d_ws, size_t ws_size,
                              hipStream_t stream) {
  (void)in_sizes; (void)n_in; (void)out_size; (void)ws_size;
  const float* xs   = (const float*)d_in[0];
  const float* w12  = (const float*)d_in[1];
  const float* w24  = (const float*)d_in[2];
  const float* w48  = (const float*)d_in[3];
  const float* sb   = (const float*)d_in[4];
  const float* a1iw = (const float*)d_in[5];
  const float* a1ib = (const float*)d_in[6];
  const float* a1ow = (const float*)d_in[7];
  const float* a1ob = (const float*)d_in[8];
  const float* a1lg = (const float*)d_in[9];
  const float* a1lb = (const float*)d_in[10];
  const float* a2iw = (const float*)d_in[11];
  const float* a2ib = (const float*)d_in[12];
  const float* a2ow = (const float*)d_in[13];
  const float* a2ob = (const float*)d_in[14];
  const float* a2lg = (const float*)d_in[15];
  const float* a2lb = (const float*)d_in[16];
  const float* fw1  = (const float*)d_in[17];
  const float* fb1  = (const float*)d_in[18];
  const float* fw2  = (const float*)d_in[19];
  const float* fb2  = (const float*)d_in[20];
  float* out = (float*)d_out;

  // workspace layout (floats)
  float* ws = (float*)d_ws;
  size_t o = 0;
  float* XINV = ws + o;      o += (size_t)Bn * Cn * Tn;        // 8388608
  int*   PER  = (int*)(ws + o); o += 1024;                     // periods [C,2]
  float* P512 = ws + o;      o += 48 * 512;                    // pinvT trend
  float* A1   = ws + o;      o += (size_t)Bn * Cn * 48;        // anchor proj
  float* COMB = ws + o;      o += (size_t)Bn * Cn * 3 * Dm;    // [s1|s2|g]
  float* Hbuf = ws + o;      o += (size_t)Bn * Cn * Dm;        // fusion hidden
  float* WKEFF = ws + o;     o += 6 * 512 * 48;
  float* WVEFF = ws + o;     o += 6 * 512 * 48;
  float* WQK  = ws + o;      o += (size_t)6 * 512 * 192;
  float* BQK  = ws + o;      o += 6 * 192;
  float* WCTX = ws + o;      o += (size_t)6 * 192 * 512;
  float* BOUT = ws + o;      o += 2 * 512;
  float* BVEFF = ws + o;     o += 2 * 512;

  // 1) transpose x_s -> x_inv
  k_transpose<<<dim3(Tn / 32, Cn / 32, Bn), dim3(32, 8), 0, stream>>>(xs, XINV);

  // 2) ACF periods
  k_acf<<<dim3(Cn), dim3(256), 0, stream>>>(XINV, PER);

  // 3) trend pinv (p=512, a=48)
  k_pinv512<<<dim3(1), dim3(256), (48 * 96 + 64) * sizeof(float), stream>>>(P512);

  // 4) A1 = x_inv @ pinv512  (M=16384, N=48, K=512)
  k_gemm<<<dim3(256, 1), dim3(256), 0, stream>>>(XINV, Tn, P512, 512, nullptr,
                                                 A1, 48, Bn * Cn, 48, 512, 0);
  // 5) g_stream = A1 @ w48^T + sb  -> COMB[:, 1024:1536]
  k_gemm<<<dim3(256, 8), dim3(256), 0, stream>>>(A1, 48, w48, 48, sb,
                                                 COMB + 2 * Dm, 3 * Dm,
                                                 Bn * Cn, Dm, 48, 0);

  // 6) fold attention weights
  k_weff<<<dim3((2 * 3 * 512 * 48 + 255) / 256), dim3(256), 0, stream>>>(
      a1iw, a2iw, w12, w24, w48, WKEFF, WVEFF);
  k_bveff<<<dim3(4), dim3(256), 0, stream>>>(sb, a1iw, a2iw, a1ib, a2ib, BVEFF);
  k_wqk_wctx<<<dim3((2 * 3 * 512 * 192 + 255) / 256), dim3(256), 0, stream>>>(
      a1iw, a2iw, a1ow, a2ow, WKEFF, WVEFF, WQK, WCTX);
  k_bqk_bout<<<dim3((2 * 3 * 192 + 2 * 512 + 255) / 256), dim3(256), 0, stream>>>(
      a1ib, a2ib, a1ow, a2ow, a1ob, a2ob, WKEFF, BVEFF, BQK, BOUT);

  // 7) aggregation -> COMB[:, 0:1024]
  k_aggregate<<<dim3(Cn, 2), dim3(256), 38144 * sizeof(float), stream>>>(
      XINV, PER, w12, w24, w48, sb, WQK, BQK, WCTX, BOUT,
      a1lg, a1lb, a2lg, a2lb, COMB);

  // 8) fusion MLP
  k_gemm<<<dim3(256, 8), dim3(256), 0, stream>>>(COMB, 3 * Dm, fw1, 3 * Dm, fb1,
                                                 Hbuf, Dm, Bn * Cn, Dm, 3 * Dm, 1);
  k_gemm<<<dim3(256, 8), dim3(256), 0, stream>>>(Hbuf, Dm, fw2, Dm, fb2,
                                                 out, Dm, Bn * Cn, Dm, Dm, 0);
}